// RMSA_77541339562176
// MI455X (gfx1250) — compile-verified
//
#include <hip/hip_runtime.h>
#include <math.h>

typedef __attribute__((ext_vector_type(2))) float v2f;
typedef __attribute__((ext_vector_type(4))) float v4f;
typedef __attribute__((ext_vector_type(8))) float v8f;

#define B_ 2
#define C_ 64
#define H_ 256
#define W_ 256
#define S_ (H_ * W_)
#define KK 576            // C*9 contraction length for the deform conv GEMM
#define HEADS 4
#define CH 16             // C / HEADS
#define NP 32             // pixels (N) per block in the conv kernel

// workspace layout (in floats)
#define CONV_OFF  0
#define NORM_OFF  (B_ * C_ * S_)        // 256 floats: [0..127]=|q| rows, [128..255]=|kv| rows
#define GRAM_OFF  (NORM_OFF + 256)      // 8 * 16 * 16
#define ATTN_OFF  (GRAM_OFF + 2048)     // 8 * 16 * 16

// ---------------------------------------------------------------------------
// Kernel 1: flow-guided deformable conv as WMMA GEMM  W[64x576] * S[576x32]
// One block = 32 consecutive pixels of one image row; 4 waves = 4 M-tiles,
// each wave computes two 16x16 WMMA tiles (N=32) reusing one A fragment.
// LDS B layout interleaved by K-pairs: s[(K>>1)*64 + p*2 + (K&1)] so each
// lane's B fragment is one 8-byte ds_load_b64.
// ---------------------------------------------------------------------------
__global__ __launch_bounds__(128)
void deform_conv_wmma(const float* __restrict__ clone,
                      const float* __restrict__ u,
                      const float* __restrict__ v,
                      const float* __restrict__ wgt,
                      float* __restrict__ conv)
{
    __shared__ float s_lds[KK * NP];           // 72 KB
    __shared__ int   p_y0[9 * NP], p_x0[9 * NP];
    __shared__ float p_wy[9 * NP], p_wx[9 * NP];

    const int tid   = threadIdx.x;
    const int tile  = blockIdx.x;              // B * H * (W/32) = 4096
    const int wtile = tile & 7;
    const int h     = (tile >> 3) & 255;
    const int b     = tile >> 11;
    const int w0    = wtile * NP;

    // stage 1: bilinear setup per (tap k, pixel p) -- shared across channels
    for (int j = tid; j < 9 * NP; j += 128) {
        int p = j & (NP - 1);
        int k = j >> 5;                        // 0..8
        int wx_pix = w0 + p;
        float uv = u[(size_t)b * S_ + h * W_ + wx_pix];
        float vv = v[(size_t)b * S_ + h * W_ + wx_pix];
        float py = (float)h      + (float)(k / 3 - 1) + vv;
        float px = (float)wx_pix + (float)(k % 3 - 1) + uv;
        float y0f = floorf(py), x0f = floorf(px);
        p_y0[j] = (int)y0f;  p_x0[j] = (int)x0f;
        p_wy[j] = py - y0f;  p_wx[j] = px - x0f;
    }
    __syncthreads();

    // stage 2: gather into interleaved B layout; kc = c*9 + k matches weight
    for (int i = tid; i < KK * NP; i += 128) {
        int p  = i & (NP - 1);
        int kc = i >> 5;
        int k  = kc % 9;
        int c  = kc / 9;
        int j  = k * NP + p;
        int y0 = p_y0[j], x0 = p_x0[j];
        int y1 = y0 + 1,  x1 = x0 + 1;
        float wy = p_wy[j], wx = p_wx[j];
        const float* img = clone + ((size_t)(b * C_ + c)) * S_;
        float s00 = 0.f, s01 = 0.f, s10 = 0.f, s11 = 0.f;
        if (y0 >= 0 && y0 < H_) {
            if (x0 >= 0 && x0 < W_) s00 = img[y0 * W_ + x0];
            if (x1 >= 0 && x1 < W_) s01 = img[y0 * W_ + x1];
        }
        if (y1 >= 0 && y1 < H_) {
            if (x0 >= 0 && x0 < W_) s10 = img[y1 * W_ + x0];
            if (x1 >= 0 && x1 < W_) s11 = img[y1 * W_ + x1];
        }
        float sval = s00 * (1.f - wy) * (1.f - wx) + s01 * (1.f - wy) * wx
                   + s10 * wy        * (1.f - wx) + s11 * wy        * wx;
        s_lds[(kc >> 1) * (2 * NP) + p * 2 + (kc & 1)] = sval;
    }
    __syncthreads();

    // stage 3: 144 K-steps, 2 x V_WMMA_F32_16X16X4_F32 per step (uniform flow)
    const int wave = tid >> 5;
    const int lane = tid & 31;
    const int row  = lane & 15;
    const int ph   = lane >> 4;                // 0: K=0,1  1: K=2,3 half
    const int om   = wave * 16;                // output-channel tile base

    v8f acc0 = {}, acc1 = {};
    const v2f* wrow2 = (const v2f*)(wgt + (size_t)(om + row) * KK);
    const v2f* s2    = (const v2f*)s_lds;
    #pragma unroll 2
    for (int step = 0; step < 144; ++step) {
        int pidx = step * 2 + ph;              // K-pair index
        v2f a  = wrow2[pidx];                  // A[M=row][K=2*pidx, +1]
        v2f b0 = s2[pidx * NP + row];          // B[K][N=row],      N-tile 0
        v2f b1 = s2[pidx * NP + 16 + row];     // B[K][N=16+row],   N-tile 1
        acc0 = __builtin_amdgcn_wmma_f32_16x16x4_f32(false, a, false, b0,
                                                     (short)0, acc0, false, false);
        acc1 = __builtin_amdgcn_wmma_f32_16x16x4_f32(false, a, false, b1,
                                                     (short)0, acc1, false, false);
    }

    // C/D layout: VGPR r -> (M=r, N=lane) lanes 0-15; (M=8+r, N=lane-16) 16-31
    const int mhi = ph * 8;
    float* o0 = conv + ((size_t)(b * C_ + om + mhi)) * S_ + h * W_ + w0 + row;
    #pragma unroll
    for (int r = 0; r < 8; ++r) {
        o0[(size_t)r * S_]      = acc0[r];
        o0[(size_t)r * S_ + 16] = acc1[r];
    }
}

// ---------------------------------------------------------------------------
// Kernel 2: L2 norms of the 128 q rows (x) and 128 kv rows (conv)
// ---------------------------------------------------------------------------
__global__ __launch_bounds__(256)
void row_norms(const float* __restrict__ x,
               const float* __restrict__ conv,
               float* __restrict__ norms)
{
    __shared__ float red[256];
    const int r = blockIdx.x;                  // 0..255
    const float* src = (r < 128) ? (x + (size_t)r * S_)
                                 : (conv + (size_t)(r - 128) * S_);
    const v4f* src4 = (const v4f*)src;
    float acc = 0.f;
    for (int i = threadIdx.x; i < S_ / 4; i += 256) {
        v4f t = src4[i];
        acc += t.x * t.x + t.y * t.y + t.z * t.z + t.w * t.w;
    }
    red[threadIdx.x] = acc;
    __syncthreads();
    for (int s = 128; s > 0; s >>= 1) {
        if (threadIdx.x < s) red[threadIdx.x] += red[threadIdx.x + s];
        __syncthreads();
    }
    if (threadIdx.x == 0) norms[r] = fmaxf(sqrtf(red[0]), 1e-12f);
}

// ---------------------------------------------------------------------------
// Kernel 3a: zero the Gram accumulator (ws is poisoned by the harness)
// ---------------------------------------------------------------------------
__global__ void zero_gram(float* __restrict__ g)
{
    g[blockIdx.x * 256 + threadIdx.x] = 0.f;
}

// ---------------------------------------------------------------------------
// Kernel 3b: 16x16 Gram per (b,head), K=65536 sliced across waves (WMMA)
// ---------------------------------------------------------------------------
#define GRAM_SLICES 32
__global__ __launch_bounds__(128)
void gram_wmma(const float* __restrict__ x,
               const float* __restrict__ conv,
               float* __restrict__ gram)
{
    const int bh    = blockIdx.x / GRAM_SLICES;     // 0..7
    const int slice = blockIdx.x % GRAM_SLICES;
    const int wave  = threadIdx.x >> 5;
    const int lane  = threadIdx.x & 31;
    const int row   = lane & 15;
    const int sh    = (lane >> 4) << 1;

    const int b  = bh / HEADS;
    const int hh = bh % HEADS;
    const float* q  = x    + ((size_t)(b * C_ + hh * CH + row)) * S_;  // A: M=row
    const float* kv = conv + ((size_t)(b * C_ + hh * CH + row)) * S_;  // B: N=row

    const int span = S_ / (GRAM_SLICES * 4);        // 512 -> 128 WMMA steps
    const int s0   = (slice * 4 + wave) * span;

    v8f acc = {};
    for (int s = s0; s < s0 + span; s += 4) {
        v2f a  = *(const v2f*)(q  + s + sh);        // 8B-aligned pair loads
        v2f bb = *(const v2f*)(kv + s + sh);
        acc = __builtin_amdgcn_wmma_f32_16x16x4_f32(false, a, false, bb,
                                                    (short)0, acc, false, false);
    }

    float* g = gram + bh * 256;
    const int mhi = (lane >> 4) * 8;
    #pragma unroll
    for (int r = 0; r < 8; ++r)
        atomicAdd(&g[(mhi + r) * 16 + row], acc[r]);
}

// ---------------------------------------------------------------------------
// Kernel 4: normalize Gram by row norms, scale by temperature, softmax rows
// ---------------------------------------------------------------------------
__global__ __launch_bounds__(128)
void attn_softmax(const float* __restrict__ gram,
                  const float* __restrict__ norms,
                  const float* __restrict__ temp,
                  float* __restrict__ attn)
{
    const int t  = threadIdx.x;                // 0..127 -> (bh, c)
    const int bh = t >> 4;
    const int c  = t & 15;
    const int b  = bh / HEADS;
    const int hh = bh % HEADS;

    const float qn = norms[b * C_ + hh * CH + c];
    const float T  = temp[hh];
    const float* g = gram + bh * 256 + c * 16;

    float l[16], m = -1e30f;
    #pragma unroll
    for (int d = 0; d < 16; ++d) {
        float kn = norms[128 + b * C_ + hh * CH + d];
        l[d] = g[d] / (qn * kn) * T;
        m = fmaxf(m, l[d]);
    }
    float sum = 0.f;
    #pragma unroll
    for (int d = 0; d < 16; ++d) { l[d] = __expf(l[d] - m); sum += l[d]; }
    const float inv = 1.f / sum;
    float* o = attn + bh * 256 + c * 16;
    #pragma unroll
    for (int d = 0; d < 16; ++d) o[d] = l[d] * inv;
}

// ---------------------------------------------------------------------------
// Kernel 5: out[c][s] = attn[16x16] * kv[16 x S]  (K=16 -> 4 WMMA steps)
// ---------------------------------------------------------------------------
__global__ __launch_bounds__(128)
void out_wmma(const float* __restrict__ conv,
              const float* __restrict__ attn,
              float* __restrict__ out)
{
    const int wave  = threadIdx.x >> 5;
    const int lane  = threadIdx.x & 31;
    const int gidx  = blockIdx.x * 4 + wave;   // 8 bh * 4096 s-tiles
    const int bh    = gidx >> 12;
    const int stile = gidx & 4095;
    const int sbase = stile * 16;
    const int b     = bh / HEADS;
    const int hh    = bh % HEADS;

    const float* A   = attn + bh * 256;
    const float* kv0 = conv + ((size_t)(b * C_ + hh * CH)) * S_ + sbase;

    const int row = lane & 15;
    const int kh  = (lane >> 4) << 1;

    v8f acc = {};
    #pragma unroll
    for (int step = 0; step < 4; ++step) {
        int k0 = step * 4;
        v2f a = *(const v2f*)(A + row * 16 + k0 + kh);   // A[M=row][K], 8B aligned
        v2f bb;
        bb.x = kv0[(size_t)(k0 + kh) * S_ + row];        // B[K=d][N=row] (coalesced)
        bb.y = kv0[(size_t)(k0 + kh + 1) * S_ + row];
        acc = __builtin_amdgcn_wmma_f32_16x16x4_f32(false, a, false, bb,
                                                    (short)0, acc, false, false);
    }

    const int mhi = (lane >> 4) * 8;
    float* o = out + ((size_t)(b * C_ + hh * CH + mhi)) * S_ + sbase + row;
    #pragma unroll
    for (int r = 0; r < 8; ++r)
        o[(size_t)r * S_] = acc[r];
}

// ---------------------------------------------------------------------------
extern "C" void kernel_launch(void* const* d_in, const int* in_sizes, int n_in,
                              void* d_out, int out_size, void* d_ws, size_t ws_size,
                              hipStream_t stream)
{
    const float* clone = (const float*)d_in[0];
    const float* x     = (const float*)d_in[1];
    const float* u     = (const float*)d_in[2];
    const float* v     = (const float*)d_in[3];
    const float* wgt   = (const float*)d_in[4];
    const float* temp  = (const float*)d_in[5];
    float* out = (float*)d_out;
    float* ws  = (float*)d_ws;

    float* conv  = ws + CONV_OFF;
    float* norms = ws + NORM_OFF;
    float* gram  = ws + GRAM_OFF;
    float* attn  = ws + ATTN_OFF;

    deform_conv_wmma<<<B_ * H_ * (W_ / NP), 128, 0, stream>>>(clone, u, v, wgt, conv);
    row_norms<<<256, 256, 0, stream>>>(x, conv, norms);
    zero_gram<<<8, 256, 0, stream>>>(gram);
    gram_wmma<<<8 * GRAM_SLICES, 128, 0, stream>>>(x, conv, gram);
    attn_softmax<<<1, 128, 0, stream>>>(gram, norms, temp, attn);
    out_wmma<<<8 * 4096 / 4, 128, 0, stream>>>(conv, attn, out);
}